// GINLayer_76587856822961
// MI455X (gfx1250) — compile-verified
//
#include <hip/hip_runtime.h>

#define D 128
#define NNODES 50000
#define NEDGES 600000
#define MT 5            // M-tiles (of 16 rows) per block
#define ROWS (MT * 16)  // 80 rows per block; 50000/80 = 625 exactly

typedef __attribute__((ext_vector_type(2))) float v2f;
typedef __attribute__((ext_vector_type(8))) float v8f;

// ---------------------------------------------------------------------------
// Kernel 0: h = (1 + eps) * feat   (written into d_out, which doubles as the
// aggregation accumulator; 4 floats per thread, fully coalesced b128 traffic)
// ---------------------------------------------------------------------------
__global__ __launch_bounds__(256) void k_init(const float* __restrict__ feat,
                                              const float* __restrict__ eps,
                                              float* __restrict__ h) {
    const int i = (blockIdx.x * 256 + threadIdx.x) * 4;
    const float s = 1.0f + eps[0];
    const float4 f = *reinterpret_cast<const float4*>(feat + i);
    float4 o;
    o.x = s * f.x; o.y = s * f.y; o.z = s * f.z; o.w = s * f.w;
    *reinterpret_cast<float4*>(h + i) = o;
}

// ---------------------------------------------------------------------------
// Kernel 1: per-edge message relu(feat[src] + edge_emb[etype]) scatter-added
// into h[dst]. One wave32 per edge: lane handles a float4 (32*4 = 128 = D).
// unsafeAtomicAdd -> global_atomic_add_f32 resolved in L2 (acc fits in L2).
// ---------------------------------------------------------------------------
__global__ __launch_bounds__(256) void k_scatter(const float* __restrict__ feat,
                                                 const int* __restrict__ src,
                                                 const int* __restrict__ dst,
                                                 const int* __restrict__ et,
                                                 const float* __restrict__ emb,
                                                 float* __restrict__ h) {
    const int wave = threadIdx.x >> 5;
    const int lane = threadIdx.x & 31;
    const int e = blockIdx.x * 8 + wave;
    const int s = src[e];
    const int d = dst[e];
    const int t = et[e];
    const float4 fs = *reinterpret_cast<const float4*>(feat + s * D + lane * 4);
    const float4 em = *reinterpret_cast<const float4*>(emb + t * D + lane * 4);
    float* dp = h + d * D + lane * 4;
    const float m0 = fmaxf(fs.x + em.x, 0.0f);
    const float m1 = fmaxf(fs.y + em.y, 0.0f);
    const float m2 = fmaxf(fs.z + em.z, 0.0f);
    const float m3 = fmaxf(fs.w + em.w, 0.0f);
    unsafeAtomicAdd(dp + 0, m0);
    unsafeAtomicAdd(dp + 1, m1);
    unsafeAtomicAdd(dp + 2, m2);
    unsafeAtomicAdd(dp + 3, m3);
}

// ---------------------------------------------------------------------------
// Kernel 2: out = relu(h @ w1 + b1) @ w2 + b2 via V_WMMA_F32_16X16X4_F32.
// One block = 8 waves = 80 nodes (5 M-tiles), so each weight fragment loaded
// from L2 is reused across 5 WMMAs (w1+w2 traffic: 256KB per 80 rows).
// Phase 1: wave w owns hidden columns [32w, 32w+32) (two N-tiles, A reused
// across both). Phase 2: wave w owns output columns [16w, 16w+16).
// LDS strides 132/260 (== 4 mod 64): column reads hit banks 4*M (distinct for
// M=0..15) -> conflict-free, and rows stay 16B-aligned for b128 staging.
// Reads its exclusive 80 rows of h from `hout`, then overwrites with output.
// ---------------------------------------------------------------------------
__global__ __launch_bounds__(256) void k_mlp(const float* __restrict__ W1,
                                             const float* __restrict__ B1,
                                             const float* __restrict__ W2,
                                             const float* __restrict__ B2,
                                             float* __restrict__ hout) {
    __shared__ float hs[ROWS * 132];    // h tile     [80 x 128]
    __shared__ float hid[ROWS * 260];   // hidden     [80 x 256]
    const int tid = threadIdx.x;
    const int base = blockIdx.x * ROWS;

    // stage the 80x128 h tile into LDS: 2560 float4 chunks, 10 per thread
    #pragma unroll
    for (int i = 0; i < 10; ++i) {
        const int chunk = tid + i * 256;
        const int r = chunk >> 5;          // 32 float4 per row
        const int c = (chunk & 31) * 4;
        const float4 v = *reinterpret_cast<const float4*>(hout + (base + r) * D + c);
        *reinterpret_cast<float4*>(&hs[r * 132 + c]) = v;
    }
    __syncthreads();

    const int w = tid >> 5;
    const int lane = tid & 31;
    const int M = lane & 15;          // row (A) / col (B,C,D) index
    const int half = lane >> 4;       // 0: lanes 0-15, 1: lanes 16-31
    const int koff = half * 2;        // K offset per ISA f32 A/B layout

    // ---- phase 1: hidden = relu(h @ w1 + b1) ----
    {
        v8f acc0[MT] = {};            // N-tile 0: cols [32w, 32w+16)
        v8f acc1[MT] = {};            // N-tile 1: cols [32w+16, 32w+32)
        for (int k0 = 0; k0 < D; k0 += 4) {
            const float* bp = W1 + (k0 + koff) * 256 + w * 32 + M;
            v2f b0, b1;
            b0.x = bp[0];   b0.y = bp[256];
            b1.x = bp[16];  b1.y = bp[16 + 256];
            #pragma unroll
            for (int m = 0; m < MT; ++m) {
                v2f a;
                a.x = hs[(m * 16 + M) * 132 + k0 + koff];
                a.y = hs[(m * 16 + M) * 132 + k0 + koff + 1];
                acc0[m] = __builtin_amdgcn_wmma_f32_16x16x4_f32(
                              false, a, false, b0, (short)0, acc0[m], false, false);
                acc1[m] = __builtin_amdgcn_wmma_f32_16x16x4_f32(
                              false, a, false, b1, (short)0, acc1[m], false, false);
            }
        }
        const float bias0 = B1[w * 32 + M];
        const float bias1 = B1[w * 32 + 16 + M];
        #pragma unroll
        for (int m = 0; m < MT; ++m) {
            #pragma unroll
            for (int r = 0; r < 8; ++r) {
                const int row = m * 16 + r + half * 8;
                const float v0 = acc0[m][r] + bias0;
                const float v1 = acc1[m][r] + bias1;
                hid[row * 260 + w * 32 + M]      = v0 > 0.0f ? v0 : 0.0f;
                hid[row * 260 + w * 32 + 16 + M] = v1 > 0.0f ? v1 : 0.0f;
            }
        }
    }
    __syncthreads();

    // ---- phase 2: out = hidden @ w2 + b2 ----
    {
        v8f acc[MT] = {};
        for (int k0 = 0; k0 < 2 * D; k0 += 4) {
            const float* bp = W2 + (k0 + koff) * D + w * 16 + M;
            v2f b;
            b.x = bp[0];
            b.y = bp[D];
            #pragma unroll
            for (int m = 0; m < MT; ++m) {
                v2f a;
                a.x = hid[(m * 16 + M) * 260 + k0 + koff];
                a.y = hid[(m * 16 + M) * 260 + k0 + koff + 1];
                acc[m] = __builtin_amdgcn_wmma_f32_16x16x4_f32(
                             false, a, false, b, (short)0, acc[m], false, false);
            }
        }
        const float bias = B2[w * 16 + M];
        #pragma unroll
        for (int m = 0; m < MT; ++m) {
            #pragma unroll
            for (int r = 0; r < 8; ++r) {
                hout[(base + m * 16 + r + half * 8) * D + w * 16 + M] = acc[m][r] + bias;
            }
        }
    }
}

// ---------------------------------------------------------------------------
extern "C" void kernel_launch(void* const* d_in, const int* in_sizes, int n_in,
                              void* d_out, int out_size, void* d_ws, size_t ws_size,
                              hipStream_t stream) {
    const float* feat = (const float*)d_in[0];
    const int*   src  = (const int*)d_in[1];
    const int*   dst  = (const int*)d_in[2];
    const int*   et   = (const int*)d_in[3];
    const float* emb  = (const float*)d_in[4];
    const float* w1   = (const float*)d_in[5];
    const float* b1   = (const float*)d_in[6];
    const float* w2   = (const float*)d_in[7];
    const float* b2   = (const float*)d_in[8];
    const float* eps  = (const float*)d_in[9];
    float* out = (float*)d_out;

    // (NNODES*D/4)/256 = 6250, NEDGES/8 = 75000, NNODES/ROWS = 625 (all exact)
    k_init   <<<(NNODES * D / 4) / 256, 256, 0, stream>>>(feat, eps, out);
    k_scatter<<<NEDGES / 8,            256, 0, stream>>>(feat, src, dst, et, emb, out);
    k_mlp    <<<NNODES / ROWS,         256, 0, stream>>>(w1, b1, w2, b2, out);
}